// GPTNet_option_ssl_28793460752781
// MI455X (gfx1250) — compile-verified
//
#include <hip/hip_runtime.h>
#include <cstdint>
#include <cstddef>

// ---------------------------------------------------------------------------
// CDNA5 (gfx1250) fp32 WMMA helpers: V_WMMA_F32_16X16X4_F32
//   A: 16x4 f32  -> 2 VGPRs/lane (lanes 0-15: M=lane,K={0,1}; 16-31: K={2,3})
//   B: 4x16 f32  -> 2 VGPRs/lane (lanes 0-15: K={0,1};  16-31: K={2,3}; N=lane&15)
//   C/D: 16x16   -> 8 VGPRs/lane (VGPR j: M=j+8*(lane>>4), N=lane&15)
// ---------------------------------------------------------------------------
typedef float v2f __attribute__((ext_vector_type(2)));
typedef float v8f __attribute__((ext_vector_type(8)));

#define NEG_SLOPE 0.1f
#define EPSBN 1e-5f

__device__ __forceinline__ v8f wmma_f32_4(v2f a, v2f b, v8f c) {
  // 8 args: (neg_a, A, neg_b, B, c_mod, C, reuse_a, reuse_b)
  return __builtin_amdgcn_wmma_f32_16x16x4_f32(false, a, false, b, (short)0, c,
                                               false, false);
}

// gfx1250 native V_TANH_F32 (TRANS op; co-executes with WMMA). Guarded fallback.
__device__ __forceinline__ float fast_tanh(float x) {
#if __has_builtin(__builtin_amdgcn_tanhf)
  return __builtin_amdgcn_tanhf(x);
#elif __has_builtin(__builtin_amdgcn_tanh_f32)
  return __builtin_amdgcn_tanh_f32(x);
#else
  return tanhf(x);
#endif
}

__device__ __forceinline__ v8f vzero8() {
  v8f z;
#pragma unroll
  for (int i = 0; i < 8; ++i) z[i] = 0.f;
  return z;
}

__device__ __forceinline__ float leaky_f(float x) {
  return x > 0.f ? x : NEG_SLOPE * x;
}

// ---------------------------------------------------------------------------
// Generic conv1x1 / temporal-conv GEMM:
//   out[n,o,t,v] = bias[o] + sum_{c,tap} W[o, c*NTAPS+tap] * in[n, c, t+tap-pad, v]
// NTAPS=1 -> plain conv1x1 (pad=0). NTAPS=7 -> TCN along T with zero pad.
// Block: 128 threads = 4 waves, each wave owns 16 output columns, loops MT
// 16-row M-tiles. Grid: (ceil(P/64), N). O = MT*16, Kdim = Cin*NTAPS (mult of 4).
// ---------------------------------------------------------------------------
template <int NTAPS, int MT>
__global__ __launch_bounds__(128) void conv_gemm_kernel(
    const float* __restrict__ in, const float* __restrict__ W,
    const float* __restrict__ bias, float* __restrict__ out, int Cin, int T,
    int V, int pad) {
  const int P = T * V;
  const int Kdim = Cin * NTAPS;
  const int O = MT * 16;
  const int n = blockIdx.y;
  const int wave = threadIdx.x >> 5;
  const int lane = threadIdx.x & 31;
  const int half = lane >> 4;
  const int l16 = lane & 15;

  int p = blockIdx.x * 64 + wave * 16 + l16;
  if (p >= P) p = P - 1;  // launches keep P % 64 == 0; clamp is branchless safety
  const int t = p / V;
  const int v = p - t * V;
  const float* inN = in + (size_t)n * Cin * P;

  v8f acc[MT];
#pragma unroll
  for (int mt = 0; mt < MT; ++mt) acc[mt] = vzero8();

  for (int k = 0; k < Kdim; k += 4) {
    const int ck0 = k + 2 * half;
    const int ck1 = ck0 + 1;
    const int c0 = ck0 / NTAPS, tap0 = ck0 - c0 * NTAPS;
    const int c1 = ck1 / NTAPS, tap1 = ck1 - c1 * NTAPS;
    const int tr0 = t + tap0 - pad;
    const int tr1 = t + tap1 - pad;
    const int tr0c = tr0 < 0 ? 0 : (tr0 >= T ? T - 1 : tr0);
    const int tr1c = tr1 < 0 ? 0 : (tr1 >= T ? T - 1 : tr1);
    const float m0 = ((unsigned)tr0 < (unsigned)T) ? 1.f : 0.f;
    const float m1 = ((unsigned)tr1 < (unsigned)T) ? 1.f : 0.f;
    v2f b;
    b.x = inN[(size_t)c0 * P + tr0c * V + v] * m0;
    b.y = inN[(size_t)c1 * P + tr1c * V + v] * m1;
#pragma unroll
    for (int mt = 0; mt < MT; ++mt) {
      const int m = mt * 16 + l16;
      v2f a;
      a.x = W[(size_t)m * Kdim + ck0];
      a.y = W[(size_t)m * Kdim + ck1];
      acc[mt] = wmma_f32_4(a, b, acc[mt]);
    }
  }

  float* outN = out + (size_t)n * O * P;
#pragma unroll
  for (int mt = 0; mt < MT; ++mt) {
#pragma unroll
    for (int j = 0; j < 8; ++j) {
      const int m = mt * 16 + j + 8 * half;
      outN[(size_t)m * P + p] = acc[mt][j] + bias[m];
    }
  }
}

// ---------------------------------------------------------------------------
// Fused spatial attention per (n,t):
//   G_s(25x25) = tanh((q_s^T k_s)/16)*alpha_s + att0_s
//   out1[n,:,t,:] = b_outs + sum_s W_s(64x64) * ( X_nt(64x25) * G_s )
// 256 threads = 8 waves. All tiles padded to 32 (zero-padded K dims).
// ---------------------------------------------------------------------------
__global__ __launch_bounds__(256) void spatial_att_kernel(
    const float* __restrict__ x, const float* __restrict__ qk,
    const float* __restrict__ att0, const float* __restrict__ alphas,
    const float* __restrict__ Wouts, const float* __restrict__ bouts,
    float* __restrict__ out) {
  const int TV = 1600;
  const int n = blockIdx.y;
  const int t = blockIdx.x;

  __shared__ float sX[64][33];   // X_nt[c][u], u>=25 zeroed
  __shared__ float sQK[96][33];  // qk[n,ch,t,:][u], u>=25 zeroed
  __shared__ float sG[32][33];   // att_s padded
  __shared__ float sY[64][33];   // Y_s = X*G

  const int tid = threadIdx.x;
  for (int idx = tid; idx < 64 * 32; idx += 256) {
    const int c = idx >> 5, u = idx & 31;
    sX[c][u] = (u < 25) ? x[(size_t)n * 64 * TV + (size_t)c * TV + t * 25 + u] : 0.f;
  }
  for (int idx = tid; idx < 96 * 32; idx += 256) {
    const int ch = idx >> 5, u = idx & 31;
    sQK[ch][u] = (u < 25) ? qk[(size_t)n * 96 * TV + (size_t)ch * TV + t * 25 + u] : 0.f;
  }
  __syncthreads();

  const int wave = tid >> 5, lane = tid & 31, half = lane >> 4, l16 = lane & 15;
  const int mt_o = wave & 3, nt_o = wave >> 2;  // out tile: M=o rows, N=v cols

  v8f accO = vzero8();

  for (int s = 0; s < 3; ++s) {
    if (wave < 4) {  // wave-uniform branch: EXEC stays all-ones inside
      const int gm = wave & 1, gn = wave >> 1;
      v8f g = vzero8();
#pragma unroll
      for (int k = 0; k < 16; k += 4) {
        v2f a, b;
        a.x = sQK[s * 16 + k + 2 * half + 0][gm * 16 + l16];  // A[u][c]=q[c][u]
        a.y = sQK[s * 16 + k + 2 * half + 1][gm * 16 + l16];
        b.x = sQK[48 + s * 16 + k + 2 * half + 0][gn * 16 + l16];
        b.y = sQK[48 + s * 16 + k + 2 * half + 1][gn * 16 + l16];
        g = wmma_f32_4(a, b, g);
      }
      const float al = alphas[s];
#pragma unroll
      for (int j = 0; j < 8; ++j) {
        const int u = gm * 16 + j + 8 * half;
        const int vv = gn * 16 + l16;
        const int uc = u < 25 ? u : 24, vc = vv < 25 ? vv : 24;
        sG[u][vv] =
            att0[s * 625 + uc * 25 + vc] + fast_tanh(g[j] * (1.f / 16.f)) * al;
      }
    }
    __syncthreads();

    // Y = X(64 x u32) * G(u32 x v32); X cols u>=25 are zero so G padding is inert
    {
      v8f y = vzero8();
#pragma unroll
      for (int k = 0; k < 32; k += 4) {
        v2f a, b;
        a.x = sX[mt_o * 16 + l16][k + 2 * half + 0];
        a.y = sX[mt_o * 16 + l16][k + 2 * half + 1];
        b.x = sG[k + 2 * half + 0][nt_o * 16 + l16];
        b.y = sG[k + 2 * half + 1][nt_o * 16 + l16];
        y = wmma_f32_4(a, b, y);
      }
#pragma unroll
      for (int j = 0; j < 8; ++j)
        sY[mt_o * 16 + j + 8 * half][nt_o * 16 + l16] = y[j];
    }
    __syncthreads();

    // accO += W_s(64x64) * Y(64 x v32)
    for (int k = 0; k < 64; k += 4) {
      v2f a, b;
      const int mrow = mt_o * 16 + l16;
      a.x = Wouts[(size_t)mrow * 192 + s * 64 + k + 2 * half + 0];
      a.y = Wouts[(size_t)mrow * 192 + s * 64 + k + 2 * half + 1];
      b.x = sY[k + 2 * half + 0][nt_o * 16 + l16];
      b.y = sY[k + 2 * half + 1][nt_o * 16 + l16];
      accO = wmma_f32_4(a, b, accO);
    }
    __syncthreads();
  }

#pragma unroll
  for (int j = 0; j < 8; ++j) {
    const int m = mt_o * 16 + j + 8 * half;
    const int vv = nt_o * 16 + l16;
    if (vv < 25)
      out[(size_t)n * 64 * TV + (size_t)m * TV + t * 25 + vv] = accO[j] + bouts[m];
  }
}

// ---------------------------------------------------------------------------
// Mean over V: xbar[n,c,t] = mean_v s_out[n,c,t,v]
// ---------------------------------------------------------------------------
__global__ __launch_bounds__(256) void mean_v_kernel(const float* __restrict__ in,
                                                     float* __restrict__ out) {
  const int i = blockIdx.x * 256 + threadIdx.x;
  if (i >= 128 * 64 * 64) return;
  const int n = i >> 12, r = i & 4095, c = r >> 6, t = r & 63;
  const float* p = in + (size_t)n * 64 * 1600 + (size_t)c * 1600 + t * 25;
  float s = 0.f;
#pragma unroll
  for (int v = 0; v < 25; ++v) s += p[v];
  out[i] = s * (1.f / 25.f);
}

// ---------------------------------------------------------------------------
// Temporal attention maps:  att[n,g,t,q], g = dir*2+s (dir0=f, dir1=b)
//   G = Q^T(64x16) x K(16x64); tanh(G/16)*alpha * causal mask
// Grid (4, N), 256 threads = 8 waves, 2 tiles/wave.
// ---------------------------------------------------------------------------
__global__ __launch_bounds__(256) void temporal_att_kernel(
    const float* __restrict__ qkt, const float* __restrict__ alphaf,
    const float* __restrict__ alphab, float* __restrict__ attt) {
  const int n = blockIdx.y;
  const int g = blockIdx.x;
  const int dir = g >> 1, s = g & 1;
  const int qg = (dir == 0) ? s : 2 + s;
  const int kg = (dir == 0) ? 4 + s : 6 + s;
  const float al = (dir == 0) ? alphaf[s] : alphab[s];
  const int wave = threadIdx.x >> 5, lane = threadIdx.x & 31;
  const int half = lane >> 4, l16 = lane & 15;
  const int mt = wave & 3, ntp = wave >> 2;
  const float* q = qkt + (size_t)n * 8192 + qg * 16 * 64;
  const float* kk = qkt + (size_t)n * 8192 + kg * 16 * 64;
  float* o = attt + (size_t)n * 16384 + g * 4096;

#pragma unroll
  for (int sub = 0; sub < 2; ++sub) {
    const int nt = ntp * 2 + sub;
    v8f gacc = vzero8();
#pragma unroll
    for (int k = 0; k < 16; k += 4) {
      v2f a, b;
      a.x = q[(k + 2 * half + 0) * 64 + mt * 16 + l16];
      a.y = q[(k + 2 * half + 1) * 64 + mt * 16 + l16];
      b.x = kk[(k + 2 * half + 0) * 64 + nt * 16 + l16];
      b.y = kk[(k + 2 * half + 1) * 64 + nt * 16 + l16];
      gacc = wmma_f32_4(a, b, gacc);
    }
#pragma unroll
    for (int j = 0; j < 8; ++j) {
      const int tt = mt * 16 + j + 8 * half;
      const int qq = nt * 16 + l16;
      const bool keep = (dir == 0) ? (tt >= qq) : (qq >= tt);
      o[tt * 64 + qq] = keep ? fast_tanh(gacc[j] * (1.f / 16.f)) * al : 0.f;
    }
  }
}

// ---------------------------------------------------------------------------
// Fused temporal mixing + W_outt conv per (n,v):
//   out2[n,:,q,v] = b_outt + sum_g W_g(64x64) * ( X_v(64x64 over c,t) * A_g(t,q) )
// Grid (25, N), 256 threads = 8 waves, 2 out tiles/wave.
// ---------------------------------------------------------------------------
__global__ __launch_bounds__(256) void temporal_apply_kernel(
    const float* __restrict__ tin, const float* __restrict__ attt,
    const float* __restrict__ Woutt, const float* __restrict__ boutt,
    float* __restrict__ out) {
  const int TV = 1600;
  const int n = blockIdx.y;
  const int v = blockIdx.x;

  __shared__ float sX[64][65];  // X_v[c][t]
  __shared__ float sY[64][65];  // Y_g[c][q]

  const int tid = threadIdx.x;
  for (int idx = tid; idx < 4096; idx += 256) {
    const int c = idx >> 6, t = idx & 63;
    sX[c][t] = tin[(size_t)n * 64 * TV + (size_t)c * TV + t * 25 + v];
  }
  __syncthreads();

  const int wave = tid >> 5, lane = tid & 31, half = lane >> 4, l16 = lane & 15;
  const int mt = wave & 3, ntp = wave >> 2;

  v8f acc2[2];
  acc2[0] = vzero8();
  acc2[1] = vzero8();

  for (int g = 0; g < 4; ++g) {
    const float* A = attt + (size_t)n * 16384 + g * 4096;
#pragma unroll
    for (int sub = 0; sub < 2; ++sub) {
      const int nt = ntp * 2 + sub;
      v8f y = vzero8();
      for (int k = 0; k < 64; k += 4) {
        v2f a, b;
        a.x = sX[mt * 16 + l16][k + 2 * half + 0];
        a.y = sX[mt * 16 + l16][k + 2 * half + 1];
        b.x = A[(k + 2 * half + 0) * 64 + nt * 16 + l16];
        b.y = A[(k + 2 * half + 1) * 64 + nt * 16 + l16];
        y = wmma_f32_4(a, b, y);
      }
#pragma unroll
      for (int j = 0; j < 8; ++j)
        sY[mt * 16 + j + 8 * half][nt * 16 + l16] = y[j];
    }
    __syncthreads();
#pragma unroll
    for (int sub = 0; sub < 2; ++sub) {
      const int nt = ntp * 2 + sub;
      for (int k = 0; k < 64; k += 4) {
        v2f a, b;
        const int mrow = mt * 16 + l16;
        a.x = Woutt[(size_t)mrow * 256 + g * 64 + k + 2 * half + 0];
        a.y = Woutt[(size_t)mrow * 256 + g * 64 + k + 2 * half + 1];
        b.x = sY[k + 2 * half + 0][nt * 16 + l16];
        b.y = sY[k + 2 * half + 1][nt * 16 + l16];
        acc2[sub] = wmma_f32_4(a, b, acc2[sub]);
      }
    }
    __syncthreads();
  }

#pragma unroll
  for (int sub = 0; sub < 2; ++sub) {
#pragma unroll
    for (int j = 0; j < 8; ++j) {
      const int m = mt * 16 + j + 8 * half;
      const int qq = (ntp * 2 + sub) * 16 + l16;
      out[(size_t)n * 64 * TV + (size_t)m * TV + qq * 25 + v] =
          acc2[sub][j] + boutt[m];
    }
  }
}

// ---------------------------------------------------------------------------
// BatchNorm: deterministic two-stage stats (no float atomics), then fused
// scale/shift + residual + leaky-ReLU apply. Layout (N=128, C=64, TV=1600).
// ---------------------------------------------------------------------------
__global__ __launch_bounds__(256) void bn_stats_partial(
    const float* __restrict__ in, float* __restrict__ partial) {
  const int o = blockIdx.y;
  const int tid = threadIdx.x;
  const int NP = 128 * 1600;
  float s = 0.f, sq = 0.f;
  for (int i = blockIdx.x * 256 + tid; i < NP; i += gridDim.x * 256) {
    const int n = i / 1600, p = i - n * 1600;
    const float val = in[(size_t)n * 64 * 1600 + (size_t)o * 1600 + p];
    s += val;
    sq += val * val;
  }
  __shared__ float rs[256], rq[256];
  rs[tid] = s;
  rq[tid] = sq;
  __syncthreads();
  for (int off = 128; off > 0; off >>= 1) {
    if (tid < off) {
      rs[tid] += rs[tid + off];
      rq[tid] += rq[tid + off];
    }
    __syncthreads();
  }
  if (tid == 0) {
    partial[(blockIdx.x * 64 + o) * 2 + 0] = rs[0];
    partial[(blockIdx.x * 64 + o) * 2 + 1] = rq[0];
  }
}

__global__ void bn_finalize(const float* __restrict__ partial,
                            const float* __restrict__ gamma,
                            const float* __restrict__ beta,
                            float* __restrict__ scale,
                            float* __restrict__ shift) {
  const int o = threadIdx.x;  // 64 threads
  float s = 0.f, sq = 0.f;
  for (int c = 0; c < 64; ++c) {
    s += partial[(c * 64 + o) * 2 + 0];
    sq += partial[(c * 64 + o) * 2 + 1];
  }
  const float inv = 1.f / 204800.f;
  const float mean = s * inv;
  const float var = sq * inv - mean * mean;
  const float sc = gamma[o] * rsqrtf(var + EPSBN);
  scale[o] = sc;
  shift[o] = beta[o] - mean * sc;
}

__global__ __launch_bounds__(256) void bn_apply_res_leaky(
    const float* __restrict__ in, const float* __restrict__ res,
    const float* __restrict__ scale, const float* __restrict__ shift,
    float* __restrict__ out) {
  const int i = blockIdx.x * 256 + threadIdx.x;
  if (i >= 128 * 64 * 1600) return;
  const int ch = (i / 1600) & 63;
  const float t = res[i] + scale[ch] * in[i] + shift[ch];
  out[i] = leaky_f(t);
}

// ---------------------------------------------------------------------------
// Host-side pipeline.
// ---------------------------------------------------------------------------
extern "C" void kernel_launch(void* const* d_in, const int* in_sizes, int n_in,
                              void* d_out, int out_size, void* d_ws,
                              size_t ws_size, hipStream_t stream) {
  const float* x = (const float*)d_in[0];
  const float* att0s = (const float*)d_in[1];
  const float* alphas = (const float*)d_in[2];
  const float* W_qk_s = (const float*)d_in[3];
  const float* b_qk_s = (const float*)d_in[4];
  const float* W_outs = (const float*)d_in[5];
  const float* b_outs = (const float*)d_in[6];
  const float* g_outs = (const float*)d_in[7];
  const float* be_outs = (const float*)d_in[8];
  const float* W_ffs = (const float*)d_in[9];
  const float* b_ffs = (const float*)d_in[10];
  const float* g_ffs = (const float*)d_in[11];
  const float* be_ffs = (const float*)d_in[12];
  const float* W_qk_t = (const float*)d_in[13];
  const float* b_qk_t = (const float*)d_in[14];
  const float* alphat_f = (const float*)d_in[15];
  const float* alphat_b = (const float*)d_in[16];
  const float* W_outt = (const float*)d_in[17];
  const float* b_outt = (const float*)d_in[18];
  const float* g_outt = (const float*)d_in[19];
  const float* be_outt = (const float*)d_in[20];
  const float* W_fft = (const float*)d_in[21];
  const float* b_fft = (const float*)d_in[22];
  const float* g_fft = (const float*)d_in[23];
  const float* be_fft = (const float*)d_in[24];
  const float* W_tcn = (const float*)d_in[25];
  const float* b_tcn = (const float*)d_in[26];
  const float* g_tcn = (const float*)d_in[27];
  const float* be_tcn = (const float*)d_in[28];

  float* ws = (float*)d_ws;
  size_t off = 0;
  float* qk = ws + off;   off += (size_t)128 * 96 * 1600;   // 78.6 MB
  float* bufA = ws + off; off += (size_t)128 * 64 * 1600;   // 52.4 MB
  float* bufB = ws + off; off += (size_t)128 * 64 * 1600;   // 52.4 MB
  float* sout = ws + off; off += (size_t)128 * 64 * 1600;   // 52.4 MB
  float* xbar = ws + off; off += (size_t)128 * 64 * 64;
  float* qkt = ws + off;  off += (size_t)128 * 128 * 64;
  float* attt = ws + off; off += (size_t)128 * 4 * 64 * 64;
  float* part = ws + off; off += 64 * 64 * 2;
  float* scA = ws + off;  off += 64;
  float* shA = ws + off;  off += 64;
  float* outF = (float*)d_out;

  const dim3 b128(128), b256(256);
  const int APPLY_BLKS = (128 * 64 * 1600) / 256;  // 51200

  // 1. qk = conv1x1(x, W_qk_s) : O=96, Cin=64
  conv_gemm_kernel<1, 6><<<dim3(25, 128), b128, 0, stream>>>(
      x, W_qk_s, b_qk_s, qk, 64, 64, 25, 0);
  // 2. fused spatial attention + W_outs conv -> bufA
  spatial_att_kernel<<<dim3(64, 128), b256, 0, stream>>>(
      x, qk, att0s, alphas, W_outs, b_outs, bufA);
  // 3. BN1; s1 = leaky(x + bn(bufA)) -> bufB
  bn_stats_partial<<<dim3(64, 64), b256, 0, stream>>>(bufA, part);
  bn_finalize<<<1, 64, 0, stream>>>(part, g_outs, be_outs, scA, shA);
  bn_apply_res_leaky<<<APPLY_BLKS, b256, 0, stream>>>(bufA, x, scA, shA, bufB);
  // 4. y2 = conv1x1(s1, W_ffs) -> bufA
  conv_gemm_kernel<1, 4><<<dim3(25, 128), b128, 0, stream>>>(
      bufB, W_ffs, b_ffs, bufA, 64, 64, 25, 0);
  // 5. BN2; s_out = leaky(x + bn(y2)) -> sout
  bn_stats_partial<<<dim3(64, 64), b256, 0, stream>>>(bufA, part);
  bn_finalize<<<1, 64, 0, stream>>>(part, g_ffs, be_ffs, scA, shA);
  bn_apply_res_leaky<<<APPLY_BLKS, b256, 0, stream>>>(bufA, x, scA, shA, sout);
  // 6. xbar = mean_v(s_out)
  mean_v_kernel<<<(128 * 64 * 64 + 255) / 256, b256, 0, stream>>>(sout, xbar);
  // 7. qkt[n,o,t] = W_qk_t * xbar  (O=128, Cin=64, P=64)
  conv_gemm_kernel<1, 8><<<dim3(1, 128), b128, 0, stream>>>(
      xbar, W_qk_t, b_qk_t, qkt, 64, 64, 1, 0);
  // 8. temporal attention maps (tanh + alpha + causal masks)
  temporal_att_kernel<<<dim3(4, 128), b256, 0, stream>>>(qkt, alphat_f,
                                                         alphat_b, attt);
  // 9. fused temporal mixing + W_outt conv -> bufA
  temporal_apply_kernel<<<dim3(25, 128), b256, 0, stream>>>(
      sout, attt, W_outt, b_outt, bufA);
  // 10. BN3; z1 = leaky(s_out + bn(bufA)) -> bufB
  bn_stats_partial<<<dim3(64, 64), b256, 0, stream>>>(bufA, part);
  bn_finalize<<<1, 64, 0, stream>>>(part, g_outt, be_outt, scA, shA);
  bn_apply_res_leaky<<<APPLY_BLKS, b256, 0, stream>>>(bufA, sout, scA, shA, bufB);
  // 11. y3 = conv1x1(z1, W_fft) -> bufA
  conv_gemm_kernel<1, 4><<<dim3(25, 128), b128, 0, stream>>>(
      bufB, W_fft, b_fft, bufA, 64, 64, 25, 0);
  // 12. BN4; z2 = leaky(s_out + bn(y3)) -> bufB
  bn_stats_partial<<<dim3(64, 64), b256, 0, stream>>>(bufA, part);
  bn_finalize<<<1, 64, 0, stream>>>(part, g_fft, be_fft, scA, shA);
  bn_apply_res_leaky<<<APPLY_BLKS, b256, 0, stream>>>(bufA, sout, scA, shA, bufB);
  // 13. z_tcn = 7-tap temporal conv(z2) + b_tcn -> bufA
  conv_gemm_kernel<7, 4><<<dim3(25, 128), b128, 0, stream>>>(
      bufB, W_tcn, b_tcn, bufA, 64, 64, 25, 3);
  // 14. BN5; out = leaky(z2 + bn(z_tcn)) -> d_out
  bn_stats_partial<<<dim3(64, 64), b256, 0, stream>>>(bufA, part);
  bn_finalize<<<1, 64, 0, stream>>>(part, g_tcn, be_tcn, scA, shA);
  bn_apply_res_leaky<<<APPLY_BLKS, b256, 0, stream>>>(bufA, bufB, scA, shA, outF);

  (void)in_sizes; (void)n_in; (void)out_size; (void)ws_size;
}